// YOLOD11Loss_5626407157760
// MI455X (gfx1250) — compile-verified
//
#include <hip/hip_runtime.h>

// ---------------------------------------------------------------------------
// YOLO multi-scale detection loss for MI455X (gfx1250, wave32).
//
// Roofline: only the objectness channel is dense (1/85 channels); box/cls
// channels are read only at <=512 positive cells per scale. Total traffic
// ~3.7MB vs 137MB naive -> ~0.2us at 23.3 TB/s. Reductions run through
// V_WMMA_F32_16X16X4_F32 (B = ones => row sums) in full f32 precision.
// ---------------------------------------------------------------------------

typedef __attribute__((ext_vector_type(2))) float v2f;
typedef __attribute__((ext_vector_type(8))) float v8f;

#define BATCH 16
#define NANCH 3
#define NBOX  32
#define NCLS  80
#define NCH   255            // NANCH * (5 + NCLS)

// workspace layout (float-element offsets into d_ws)
#define WS_PART 16           // per-block partials: 1200*4 + 300*4 + 75*4 = 6300
#define WS_TGT  6400         // target entries: 3 * 512 * 8 floats
#define WS_GRID 18688        // target-index grids (ints): 307200+76800+19200
#define GRID_TOTAL 403200

__device__ const float g_anch[3][3][2] = {
  {{10.f,13.f},{16.f,30.f},{33.f,23.f}},
  {{30.f,61.f},{62.f,45.f},{59.f,119.f}},
  {{116.f,90.f},{156.f,198.f},{373.f,326.f}}};
__device__ const float g_stride[3]  = {8.f, 16.f, 32.f};
__device__ const float g_balance[3] = {4.f, 1.f, 0.4f};
__device__ const int   g_hw[3]   = {80, 40, 20};
__device__ const int   g_poff[3] = {0, 4800, 6000};
__device__ const int   g_nb[3]   = {1200, 300, 75};

// ---------------------------------------------------------------------------
// helpers
// ---------------------------------------------------------------------------
__device__ __forceinline__ float bce_logits(float x, float t) {
  return fmaxf(x, 0.f) - x * t + log1pf(expf(-fabsf(x)));
}
__device__ __forceinline__ float sigmoidf(float x) {
  return 1.f / (1.f + expf(-x));
}

__device__ float ciou_f(float cx1, float cy1, float w1, float h1,
                        float cx2, float cy2, float w2, float h2) {
  const float EPSf = 1e-7f;
  float b1x1 = cx1 - 0.5f*w1, b1x2 = cx1 + 0.5f*w1;
  float b1y1 = cy1 - 0.5f*h1, b1y2 = cy1 + 0.5f*h1;
  float b2x1 = cx2 - 0.5f*w2, b2x2 = cx2 + 0.5f*w2;
  float b2y1 = cy2 - 0.5f*h2, b2y2 = cy2 + 0.5f*h2;
  float iw = fmaxf(fminf(b1x2, b2x2) - fmaxf(b1x1, b2x1), 0.f);
  float ih = fmaxf(fminf(b1y2, b2y2) - fmaxf(b1y1, b2y1), 0.f);
  float inter = iw * ih;
  float uni = w1*h1 + w2*h2 - inter + EPSf;
  float iou = inter / uni;
  float cw  = fmaxf(b1x2, b2x2) - fminf(b1x1, b2x1);
  float chh = fmaxf(b1y2, b2y2) - fminf(b1y1, b2y1);
  float c2  = cw*cw + chh*chh + EPSf;
  float dx = b2x1 + b2x2 - b1x1 - b1x2;
  float dy = b2y1 + b2y2 - b1y1 - b1y2;
  float rho2 = (dx*dx + dy*dy) * 0.25f;
  float dv = atanf(w2 / (h2 + EPSf)) - atanf(w1 / (h1 + EPSf));
  float v  = (4.0f / 9.8696044010893586f) * dv * dv;   // 4/pi^2
  float alpha = v / (v - iou + 1.0f + EPSf);
  return iou - (rho2 / c2 + v * alpha);
}

// Wave32 sum via V_WMMA_F32_16X16X4_F32: A=(v,0) per lane, B=ones.
// Row m of D = v[m] + v[m+16]; each lane's 8 D-VGPRs cover rows {0..7} or
// {8..15}; adding them + shfl_xor(16) yields the full 32-lane sum.
__device__ __forceinline__ float wave_reduce_sum(float v) {
#if __has_builtin(__builtin_amdgcn_wmma_f32_16x16x4_f32)
  v2f a; a[0] = v;   a[1] = 0.f;
  v2f b; b[0] = 1.f; b[1] = 1.f;
  v8f c = {};
  v8f d = __builtin_amdgcn_wmma_f32_16x16x4_f32(
      /*neg_a=*/false, a, /*neg_b=*/false, b,
      /*c_mod=*/(short)0, c, /*reuse_a=*/false, /*reuse_b=*/false);
  float s = d[0]+d[1]+d[2]+d[3]+d[4]+d[5]+d[6]+d[7];
  s += __shfl_xor(s, 16, 32);
  return s;
#else
  for (int o = 16; o > 0; o >>= 1) v += __shfl_xor(v, o, 32);
  return v;
#endif
}

// 256-thread block sum (8 waves), deterministic order. Result in all threads.
__device__ __forceinline__ float block_reduce_256(float v, float* lds8) {
  float w = wave_reduce_sum(v);
  int lane = threadIdx.x & 31;
  int wv   = threadIdx.x >> 5;
  __syncthreads();
  if (lane == 0) lds8[wv] = w;
  __syncthreads();
  float t = 0.f;
#pragma unroll
  for (int i = 0; i < 8; ++i) t += lds8[i];
  return t;
}

// ---------------------------------------------------------------------------
// kernels
// ---------------------------------------------------------------------------
__global__ void k_init(int* __restrict__ grid_base, int n) {
  int i = blockIdx.x * blockDim.x + threadIdx.x;
  if (i < n) grid_base[i] = -1;
}

__global__ void k_build(const float* __restrict__ boxes,
                        const int*   __restrict__ labels,
                        float* __restrict__ tent,
                        int*   __restrict__ grid_base) {
  int tid = blockIdx.x * blockDim.x + threadIdx.x;
  if (tid >= 3 * BATCH * NBOX) return;
  int s = tid / (BATCH * NBOX);
  int r = tid % (BATCH * NBOX);
  int b = r / NBOX;
  int n = r % NBOX;

  const float* bx = boxes + ((size_t)b * NBOX + n) * 4;
  float x1 = bx[0], y1 = bx[1], x2 = bx[2], y2 = bx[3];
  float cx = 0.5f * (x1 + x2), cy = 0.5f * (y1 + y2);
  float gw = x2 - x1, gh = y2 - y1;

  float stride = g_stride[s];
  int H = g_hw[s], W = H;
  float xg = cx / stride, yg = cy / stride;
  int gi = (int)yg; gi = gi < 0 ? 0 : (gi > H - 1 ? H - 1 : gi);
  int gj = (int)xg; gj = gj < 0 ? 0 : (gj > W - 1 ? W - 1 : gj);

  int best = 0; float bm = 3.4e38f;
#pragma unroll
  for (int a = 0; a < 3; ++a) {
    float aw = g_anch[s][a][0] * stride, ah = g_anch[s][a][1] * stride;
    float rw = gw / aw, rh = gh / ah;
    float m = fmaxf(rw, 1.f / rw) * fmaxf(rh, 1.f / rh);
    if (m < bm) { bm = m; best = a; }     // first-min = jnp.argmax(1/metric)
  }
  float tx = xg - (float)gj;
  float ty = yg - (float)gi;
  float aw = g_anch[s][best][0] * stride, ah = g_anch[s][best][1] * stride;
  float tw = logf(gw / (aw + 1e-16f));
  float th = logf(gh / (ah + 1e-16f));

  int e = b * NBOX + n;
  float* te = tent + ((size_t)s * BATCH * NBOX + e) * 8;
  te[0] = tx; te[1] = ty; te[2] = tw; te[3] = th;
  te[4] = (float)labels[b * NBOX + n];
  te[5] = 0.f; te[6] = 0.f; te[7] = 0.f;

  const int goff[3] = {0, 307200, 384000};
  int* grid = grid_base + goff[s];
  int cell = ((b * 3 + best) * H + gi) * W + gj;
  // deterministic collision winner: largest (b*32+n) == last writer in
  // flattened scatter order (matches JAX .at[].set last-update semantics).
  atomicMax(&grid[cell], e);
}

__global__ void k_loss(const float* __restrict__ pred,
                       const int*   __restrict__ tgrid,
                       const float* __restrict__ tent,
                       float* __restrict__ part,
                       int s) {
  __shared__ float lds8[8];
  int H = g_hw[s], W = H;
  size_t hw = (size_t)H * W;
  float stride = g_stride[s];

  int cell = blockIdx.x * 256 + threadIdx.x;     // grids are exact multiples
  int x = cell % W;
  int t = cell / W;
  int y = t % H;  t /= H;
  int a = t % 3;
  int b = t / 3;

  const float* p0 = pred + ((size_t)(b * NCH + a * 85)) * hw + (size_t)y * W + x;
  float pobj = p0[4 * hw];                        // only dense channel read
  int idx = tgrid[cell];

  float box = 0.f, obj = 0.f, cls = 0.f, npos = 0.f;
  if (idx < 0) {
    obj = 0.5f * bce_logits(pobj, 0.f);           // negatives: obj BCE only
  } else {
    obj  = bce_logits(pobj, 1.f);
    npos = 1.f;
    const float* te = tent + (size_t)idx * 8;
    float tx = te[0], ty = te[1], tw = te[2], th = te[3];
    int label = (int)te[4];
    float awx = g_anch[s][a][0] * stride, awy = g_anch[s][a][1] * stride;

    float px = p0[0], py = p0[hw], pw = p0[2 * hw], ph = p0[3 * hw];
    float pcx = (sigmoidf(px) + (float)x) * stride;
    float pcy = (sigmoidf(py) + (float)y) * stride;
    float pww = expf(pw) * awx, phh = expf(ph) * awy;
    float tcx = ((float)x + tx) * stride;
    float tcy = ((float)y + ty) * stride;
    float tww = expf(tw) * awx, thh = expf(th) * awy;
    box = 1.f - ciou_f(pcx, pcy, pww, phh, tcx, tcy, tww, thh);

    for (int c = 0; c < NCLS; ++c) {
      float pc = p0[(size_t)(5 + c) * hw];
      cls += bce_logits(pc, c == label ? 1.f : 0.f);
    }
  }

  float sb = block_reduce_256(box,  lds8);
  float so = block_reduce_256(obj,  lds8);
  float sc = block_reduce_256(cls,  lds8);
  float sn = block_reduce_256(npos, lds8);
  if (threadIdx.x == 0) {
    float* pp = part + (size_t)blockIdx.x * 4;
    pp[0] = sb; pp[1] = so; pp[2] = sc; pp[3] = sn;
  }
}

__global__ void k_final(const float* __restrict__ wsf, float* __restrict__ out) {
  __shared__ float lds8[8];
  float sums[3][4];
#pragma unroll
  for (int s = 0; s < 3; ++s) {
    const float* ps = wsf + WS_PART + g_poff[s];
    int nb = g_nb[s];
#pragma unroll
    for (int q = 0; q < 4; ++q) {
      float v = 0.f;
      for (int i = threadIdx.x; i < nb; i += 256) v += ps[i * 4 + q];
      sums[s][q] = block_reduce_256(v, lds8);    // deterministic tree
    }
  }
  if (threadIdx.x == 0) {
    float total = 0.f;
#pragma unroll
    for (int s = 0; s < 3; ++s) {
      float boxs = sums[s][0], objs = sums[s][1];
      float clss = sums[s][2], nps  = sums[s][3];
      // LAMBDA_IOU=1, LAMBDA_OBJ=1 (0.5*neg folded in), LAMBDA_CLS=0.5
      total += boxs / fmaxf(nps, 1.f) + g_balance[s] * objs + 0.5f * clss;
    }
    out[0] = total / (float)BATCH;
  }
}

// ---------------------------------------------------------------------------
// launch
// ---------------------------------------------------------------------------
extern "C" void kernel_launch(void* const* d_in, const int* in_sizes, int n_in,
                              void* d_out, int out_size, void* d_ws, size_t ws_size,
                              hipStream_t stream) {
  const float* preds[3] = {(const float*)d_in[0],
                           (const float*)d_in[1],
                           (const float*)d_in[2]};
  const float* boxes  = (const float*)d_in[3];
  const int*   labels = (const int*)d_in[4];
  float* wsf = (float*)d_ws;
  int*   wsi = (int*)d_ws;
  float* out = (float*)d_out;

  k_init<<<(GRID_TOTAL + 255) / 256, 256, 0, stream>>>(wsi + WS_GRID, GRID_TOTAL);
  k_build<<<6, 256, 0, stream>>>(boxes, labels, wsf + WS_TGT, wsi + WS_GRID);

  const int goff[3] = {0, 307200, 384000};
  const int poff[3] = {0, 4800, 6000};
  const int nblk[3] = {1200, 300, 75};
  for (int s = 0; s < 3; ++s) {
    k_loss<<<nblk[s], 256, 0, stream>>>(preds[s],
                                        wsi + WS_GRID + goff[s],
                                        wsf + WS_TGT + (size_t)s * NBOX * BATCH * 8,
                                        wsf + WS_PART + poff[s], s);
  }
  k_final<<<1, 256, 0, stream>>>(wsf, out);
}